// IsoMaxPlusLossFirstPart_29360396435989
// MI455X (gfx1250) — compile-verified
//
#include <hip/hip_runtime.h>
#include <math.h>

typedef float v2f __attribute__((ext_vector_type(2)));
typedef float v8f __attribute__((ext_vector_type(8)));

#define C 20             // channels == num prototypes
#define HW_SHIFT 19
#define HW (512*1024)    // H*W
#define BATCH 8
#define NPIX (BATCH*HW)  // 4,194,304 pixels
#define NWAVES 8
#define ITERS 8
#define PIX_PER_BLOCK (NWAVES*32*ITERS)  // 2048

__global__ __launch_bounds__(256) void isomax_wmma_kernel(
    const float* __restrict__ feat, const float* __restrict__ prot,
    const float* __restrict__ dscale, float* __restrict__ out)
{
    __shared__ float pn_lds[32*22];            // normalized prototypes, zero padded rows 20..31
    __shared__ float psq_lds[32];
    __shared__ float xn_lds[NWAVES][32*22];    // per-wave normalized pixel vectors
    __shared__ float xsq_lds[NWAVES][32];
    __shared__ float out_lds[NWAVES][C*33];    // [proto][pixel], stride 33 = conflict-free

    const int tid  = threadIdx.x;
    const int wave = tid >> 5;
    const int lane = tid & 31;
    const int lh   = lane & 15;
    const int hi   = lane >> 4;

    // ---- one-time prototype normalization (wave 0), zero-pad rows >= 20 ----
    if (wave == 0) {
        #pragma unroll
        for (int k = 0; k < 22; ++k) pn_lds[lane*22 + k] = 0.0f;
        psq_lds[lane] = 0.0f;
        if (lane < C) {
            float p[C]; float ss = 0.0f;
            #pragma unroll
            for (int k = 0; k < C; ++k) { p[k] = prot[lane*C + k]; ss = fmaf(p[k], p[k], ss); }
            float inv = __builtin_amdgcn_rcpf(fmaxf(__builtin_amdgcn_sqrtf(ss), 1e-12f));
            #pragma unroll
            for (int k = 0; k < C; ++k) pn_lds[lane*22 + k] = p[k] * inv;
            psq_lds[lane] = (ss * inv) * inv;
        }
    }
    __syncthreads();

    // ---- B fragments (prototypes^T): 2 col tiles x 5 K-chunks ----
    // ISA B layout (4x16 f32): lane = column N, lanes 0-15 hold K=kb,kb+1; lanes 16-31 hold K=kb+2,kb+3
    v2f Bf[2][5];
    float psqf[2];
    #pragma unroll
    for (int ct = 0; ct < 2; ++ct) {
        int row = lh + 16*ct;
        psqf[ct] = psq_lds[row];
        #pragma unroll
        for (int kc = 0; kc < 5; ++kc) {
            int kb = 4*kc + 2*hi;
            Bf[ct][kc].x = pn_lds[row*22 + kb];
            Bf[ct][kc].y = pn_lds[row*22 + kb + 1];
        }
    }

    const float absS = fabsf(dscale[0]);
    const float negS = -absS;

    for (int it = 0; it < ITERS; ++it) {
        const long long chunk = (long long)blockIdx.x * PIX_PER_BLOCK + (long long)(it*NWAVES + wave) * 32;
        const int b  = (int)(chunk >> HW_SHIFT);
        const int hw = (int)(chunk & (HW - 1));
        const float* fp = feat + (size_t)b * C * HW + hw + lane;

        // 20 coalesced channel loads + per-pixel L2 norm
        float x[C]; float ss = 0.0f;
        #pragma unroll
        for (int k = 0; k < C; ++k) {
            x[k] = __builtin_nontemporal_load(fp + (size_t)k * HW);
            ss = fmaf(x[k], x[k], ss);
        }
        const float inv = __builtin_amdgcn_rcpf(fmaxf(__builtin_amdgcn_sqrtf(ss), 1e-12f));
        #pragma unroll
        for (int k = 0; k < C; ++k) xn_lds[wave][lane*22 + k] = x[k] * inv;
        xsq_lds[wave][lane] = (ss * inv) * inv;

        // A fragments: lane = row M (pixel), lane halves split K (0-1 / 2-3) per ISA 16x4 f32 layout
        v2f Af[2][5];
        #pragma unroll
        for (int t = 0; t < 2; ++t) {
            int row = lh + 16*t;
            #pragma unroll
            for (int kc = 0; kc < 5; ++kc) {
                int kb = 4*kc + 2*hi;
                Af[t][kc].x = xn_lds[wave][row*22 + kb];
                Af[t][kc].y = xn_lds[wave][row*22 + kb + 1];
            }
        }

        // GEMM: D[pixel][proto] = xn . pn^T  via 20x V_WMMA_F32_16X16X4_F32
        #pragma unroll
        for (int t = 0; t < 2; ++t) {
            #pragma unroll
            for (int ct = 0; ct < 2; ++ct) {
                v8f acc = {0.f,0.f,0.f,0.f,0.f,0.f,0.f,0.f};
                #pragma unroll
                for (int kc = 0; kc < 5; ++kc) {
                    acc = __builtin_amdgcn_wmma_f32_16x16x4_f32(
                        false, Af[t][kc], false, Bf[ct][kc],
                        (short)0, acc, false, false);
                }
                const int proto = lh + 16*ct;
                #pragma unroll
                for (int v = 0; v < 8; ++v) {
                    int pix = 16*t + v + 8*hi;           // D layout: VGPR v, lane-half selects M
                    float s = xsq_lds[wave][pix] + psqf[ct];
                    float q = fmaxf(fmaf(-2.0f, acc[v], s), 0.0f);
                    float r = negS * __builtin_amdgcn_sqrtf(q);  // single v_sqrt_f32 (TRANS)
                    if (proto < C) out_lds[wave][proto*33 + pix] = r;
                }
            }
        }

        // coalesced write-out: [B, M, H, W]
        float* op = out + (size_t)b * C * HW + hw + lane;
        #pragma unroll
        for (int m = 0; m < C; ++m) {
            __builtin_nontemporal_store(out_lds[wave][m*33 + lane], op + (size_t)m * HW);
        }
    }
}

extern "C" void kernel_launch(void* const* d_in, const int* in_sizes, int n_in,
                              void* d_out, int out_size, void* d_ws, size_t ws_size,
                              hipStream_t stream) {
    (void)in_sizes; (void)n_in; (void)out_size; (void)d_ws; (void)ws_size;
    const float* feat   = (const float*)d_in[0];
    const float* prot   = (const float*)d_in[1];
    const float* dscale = (const float*)d_in[2];
    float* out = (float*)d_out;

    dim3 grid(NPIX / PIX_PER_BLOCK);  // 2048 blocks
    dim3 block(256);                  // 8 wave32
    isomax_wmma_kernel<<<grid, block, 0, stream>>>(feat, prot, dscale, out);
}